// SAM2UNetCDFSSAggressive_23940147707942
// MI455X (gfx1250) — compile-verified
//
#include <hip/hip_runtime.h>
#include <math.h>

#define EPSF 1e-6f
#define NB 8
#define DD 256
#define NN 10790

// ---- workspace float offsets ----
#define MASK_L0 0         // 8*4096
#define MASK_L1 32768     // 8*1024
#define MASK_L2 40960     // 8*256
#define CNTFG   43008     // 24  indexed (l*8+b)
#define CNTBG   43032     // 24
#define PFG     43056     // 24*256
#define PBG     49200     // 24*256
#define CNBASE  55344     // 24*512  (Cn rows: [2][256])
#define ABASE   67632     // 24*512  (A: [256][2])
#define SCORES  79920     // 8*4096
// total 112688 floats (~450 KB)

typedef __attribute__((ext_vector_type(2))) float v2f;
typedef __attribute__((ext_vector_type(8))) float v8f;

__device__ __forceinline__ int lvl_hw(int l)   { return l == 0 ? 4096 : (l == 1 ? 1024 : 256); }
__device__ __forceinline__ int lvl_mofs(int l) { return l == 0 ? MASK_L0 : (l == 1 ? MASK_L1 : MASK_L2); }

__device__ __forceinline__ float blockReduceSum(float v, float* sh) {
    int t = threadIdx.x;
    sh[t] = v; __syncthreads();
    for (int s = 128; s > 0; s >>= 1) {
        if (t < s) sh[t] += sh[t + s];
        __syncthreads();
    }
    float r = sh[0]; __syncthreads();
    return r;
}

// ---- K1: nearest-neighbor mask resize for all 3 levels ----
__global__ __launch_bounds__(256) void k_resize(const int* __restrict__ sm, float* __restrict__ ws) {
    int i = blockIdx.x * 256 + threadIdx.x;
    if (i >= 43008) return;
    int b, p, w, step, ofs;
    if (i < 32768)      { b = i >> 12;              p = i & 4095; w = 64; step = 8;  ofs = MASK_L0 + b * 4096; }
    else if (i < 40960) { int j = i - 32768; b = j >> 10; p = j & 1023; w = 32; step = 16; ofs = MASK_L1 + b * 1024; }
    else                { int j = i - 40960; b = j >> 8;  p = j & 255;  w = 16; step = 32; ofs = MASK_L2 + b * 256; }
    int y = p / w, x = p % w;
    ws[ofs + p] = (float)sm[(size_t)b * 512 * 512 + (size_t)(y * step) * 512 + x * step];
}

// ---- K2: fg/bg pixel counts per (b, level) ----
__global__ __launch_bounds__(256) void k_counts(float* __restrict__ ws) {
    __shared__ float sh[256];
    int b = blockIdx.x, l = blockIdx.y, t = threadIdx.x;
    int hw = lvl_hw(l), ofs = lvl_mofs(l) + b * hw;
    float s = 0.f;
    for (int p = t; p < hw; p += 256) s += ws[ofs + p];
    s = blockReduceSum(s, sh);
    if (t == 0) { ws[CNTFG + l * 8 + b] = s; ws[CNTBG + l * 8 + b] = (float)hw - s; }
}

// ---- K3: masked avg pools of s -> proto_fg / proto_bg ----
__global__ __launch_bounds__(256) void k_protos(const float* __restrict__ s1, const float* __restrict__ s2,
                                                const float* __restrict__ s3, float* __restrict__ ws) {
    __shared__ float sh[256];
    int c = blockIdx.x, b = blockIdx.y, l = blockIdx.z, t = threadIdx.x;
    int hw = lvl_hw(l), mofs = lvl_mofs(l) + b * hw;
    const float* S = (l == 0 ? s1 : (l == 1 ? s2 : s3)) + ((size_t)b * DD + c) * hw;
    float sf = 0.f, sb = 0.f;
    for (int p = t; p < hw; p += 256) {
        float m = ws[mofs + p], v = S[p];
        sf += m * v; sb += (1.f - m) * v;
    }
    sf = blockReduceSum(sf, sh);
    sb = blockReduceSum(sb, sh);
    if (t == 0) {
        float cf = ws[CNTFG + l * 8 + b], cb = ws[CNTBG + l * 8 + b];
        ws[PFG + (l * 8 + b) * 256 + c] = sf / (cf + EPSF);
        ws[PBG + (l * 8 + b) * 256 + c] = sb / (cb + EPSF);
    }
}

// ---- K4: Cn (normalized [bg;fg]), A = Rn^T (Cn Cn^T)^-1  (rank-2 factor of P) ----
__global__ __launch_bounds__(256) void k_build(const float* __restrict__ r1, const float* __restrict__ r2,
                                               const float* __restrict__ r3, float* __restrict__ ws) {
    __shared__ float sh[256];
    int b = blockIdx.x, l = blockIdx.y, t = threadIdx.x;
    const float* R = (l == 0 ? r1 : (l == 1 ? r2 : r3));
    int g = (l * 8 + b);
    float pf = ws[PFG + g * 256 + t];
    float pb = ws[PBG + g * 256 + t];
    float q0 = R[t], q1 = R[256 + t];
    float nb  = sqrtf(blockReduceSum(pb * pb, sh)) + EPSF;
    float nf  = sqrtf(blockReduceSum(pf * pf, sh)) + EPSF;
    float nr0 = sqrtf(blockReduceSum(q0 * q0, sh)) + EPSF;
    float nr1 = sqrtf(blockReduceSum(q1 * q1, sh)) + EPSF;
    float c0 = pb / nb, c1 = pf / nf;          // row0 = bg, row1 = fg (matches reference stack order)
    float rn0 = q0 / nr0, rn1 = q1 / nr1;
    float m00 = blockReduceSum(c0 * c0, sh);
    float m01 = blockReduceSum(c0 * c1, sh);
    float m11 = blockReduceSum(c1 * c1, sh);
    float det = m00 * m11 - m01 * m01;
    float i00 = m11 / det, i01 = -m01 / det, i11 = m00 / det;
    ws[CNBASE + g * 512 + t]       = c0;
    ws[CNBASE + g * 512 + 256 + t] = c1;
    ws[ABASE + g * 512 + t * 2]     = rn0 * i00 + rn1 * i01;
    ws[ABASE + g * 512 + t * 2 + 1] = rn0 * i01 + rn1 * i11;
}

// ---- K5: fused rank-2 transform  out = A @ (Cn @ X)  via V_WMMA_F32_16X16X4_F32 ----
// grid: (hw/32, B, 2{q,s}); block: 256 (8 waves). Writes directly into packed output.
// Level-0 s blocks also emit the per-column L2 scores for top-k (fused, saves a 33MB re-read).
__global__ __launch_bounds__(256) void k_transform(const float* __restrict__ qp, const float* __restrict__ sp,
                                                   float* __restrict__ ws, float* __restrict__ out,
                                                   int hw, int lvl, int colq, int cols) {
    __shared__ float Xs[256 * 32];
    __shared__ float Ys2[32 * 2];    // interleaved: [col][2]
    __shared__ float CnS[512];
    __shared__ float AS[512];        // [256][2]
    __shared__ float ZeroS[2];       // zero source for K=2,3 lanes
    __shared__ float colSq[32];
    int t = threadIdx.x;
    int b = blockIdx.y, isS = blockIdx.z;
    int n0 = blockIdx.x * 32;
    const float* X = (isS ? sp : qp) + (size_t)b * DD * hw;
    int g = (lvl * 8 + b) * 512;
    CnS[t]       = ws[CNBASE + g + t];
    CnS[t + 256] = ws[CNBASE + g + 256 + t];
    AS[t]        = ws[ABASE + g + t];
    AS[t + 256]  = ws[ABASE + g + 256 + t];
    if (t < 2)  ZeroS[t] = 0.f;
    if (t < 32) colSq[t] = 0.f;
    // stage X tile (256 rows x 32 cols) as float4: global_load_b128, fully coalesced
    {
        int c = t >> 3;               // 4 consecutive floats per thread
        int j = (t & 7) * 4;
        #pragma unroll
        for (int k = 0; k < 8; ++k, c += 32) {
            float4 v = *reinterpret_cast<const float4*>(X + (size_t)c * hw + n0 + j);
            *reinterpret_cast<float4*>(&Xs[c * 32 + j]) = v;
        }
    }
    __syncthreads();
    // Y(2x32) = Cn @ Xtile, stored interleaved for 8-byte B-operand loads
    if (t < 64) {
        int a = t >> 5, j = t & 31;
        const float* crow = &CnS[a * 256];
        float acc = 0.f;
        for (int c = 0; c < 256; ++c) acc += crow[c] * Xs[c * 32 + j];
        Ys2[j * 2 + a] = acc;
    }
    __syncthreads();
    // out tile (256x32) = A(256x2, K-padded to 4) @ Y(2x32): 32 WMMA tiles, 4 per wave
    int w = t >> 5, lane = t & 31, l16 = lane & 15;
    bool lo = lane < 16;
    bool doScore = (isS != 0) && (lvl == 0);
    size_t outBase = (size_t)b * DD * NN;
    int colbase = isS ? cols : colq;
    #pragma unroll
    for (int i = 0; i < 4; ++i) {
        int tt = w * 4 + i;
        int rb = tt >> 1, cb = tt & 1;       // 16 row-blocks x 2 col-blocks
        int row = rb * 16 + l16;
        int col = cb * 16 + l16;
        // address-select into a zeroed LDS region: unconditional ds_load_b64, no exec masking
        const float* ap = lo ? (AS + row * 2) : ZeroS;     // K=0,1 real / K=2,3 zero
        const float* bp = lo ? (Ys2 + col * 2) : ZeroS;
        v2f a;  a.x  = ap[0]; a.y  = ap[1];
        v2f bb; bb.x = bp[0]; bb.y = bp[1];
        v8f cc = {};
        v8f dd = __builtin_amdgcn_wmma_f32_16x16x4_f32(false, a, false, bb, (short)0, cc, false, false);
        int rbase = rb * 16 + (lo ? 0 : 8);
        int outcol = colbase + n0 + cb * 16 + l16;
        #pragma unroll
        for (int v = 0; v < 8; ++v)
            out[outBase + (size_t)(rbase + v) * NN + outcol] = dd[v];
        if (doScore) {
            float sq = 0.f;
            #pragma unroll
            for (int v = 0; v < 8; ++v) sq += dd[v] * dd[v];
            atomicAdd(&colSq[cb * 16 + l16], sq);          // ds_add_f32
        }
    }
    if (doScore) {
        __syncthreads();
        if (t < 32) ws[SCORES + b * 4096 + n0 + t] = sqrtf(colSq[t]);
    }
}

// ---- K6: masked pools of transformed s -> packed proto columns ----
__global__ __launch_bounds__(256) void k_pool_t(float* __restrict__ out, const float* __restrict__ ws) {
    __shared__ float sh[256];
    int c = blockIdx.x, b = blockIdx.y, l = blockIdx.z, t = threadIdx.x;
    int hw = lvl_hw(l), mofs = lvl_mofs(l) + b * hw;
    int scol = (l == 0 ? 5376 : (l == 1 ? 9472 : 10496));
    const float* row = out + ((size_t)b * DD + c) * NN + scol;
    float sf = 0.f, sb = 0.f;
    for (int p = t; p < hw; p += 256) {
        float m = ws[mofs + p], v = row[p];
        sf += m * v; sb += (1.f - m) * v;
    }
    sf = blockReduceSum(sf, sh);
    sb = blockReduceSum(sb, sh);
    if (t == 0) {
        float cf = ws[CNTFG + l * 8 + b], cb = ws[CNTBG + l * 8 + b];
        out[((size_t)b * DD + c) * NN + 10752 + l] = sf / (cf + EPSF);
        out[((size_t)b * DD + c) * NN + 10755 + l] = sb / (cb + EPSF);
    }
}

// ---- K7: deterministic masked top-32 (value desc, index asc) + token gather ----
__global__ __launch_bounds__(256) void k_topk(float* __restrict__ out, const float* __restrict__ ws) {
    __shared__ float sc[4096];
    __shared__ float bv[256];
    __shared__ int   bi[256];
    __shared__ int   selIdx;
    int b = blockIdx.x, t = threadIdx.x;
    float cf = ws[CNTFG + b];                          // level-0 fg count
    bool useAll = !(cf > 0.f);                         // fallback_to_full
    for (int p = t; p < 4096; p += 256) {
        float m = useAll ? 1.f : ws[MASK_L0 + b * 4096 + p];
        sc[p] = (m >= 0.5f) ? ws[SCORES + b * 4096 + p] : -INFINITY;
    }
    __syncthreads();
    for (int j = 0; j < 32; ++j) {
        float best = -INFINITY; int bidx = 0x7fffffff;
        for (int p = t * 16; p < t * 16 + 16; ++p) {
            float v = sc[p];
            if (v > best || (v == best && p < bidx)) { best = v; bidx = p; }
        }
        bv[t] = best; bi[t] = bidx; __syncthreads();
        for (int s = 128; s > 0; s >>= 1) {
            if (t < s) {
                float v2 = bv[t + s]; int i2 = bi[t + s];
                if (v2 > bv[t] || (v2 == bv[t] && i2 < bi[t])) { bv[t] = v2; bi[t] = i2; }
            }
            __syncthreads();
        }
        if (t == 0) selIdx = bi[0];
        __syncthreads();
        int idx = selIdx;
        out[((size_t)b * DD + t) * NN + 10758 + j] = out[((size_t)b * DD + t) * NN + 5376 + idx];
        if (t == 0) sc[idx] = -INFINITY;
        __syncthreads();
    }
}

extern "C" void kernel_launch(void* const* d_in, const int* in_sizes, int n_in,
                              void* d_out, int out_size, void* d_ws, size_t ws_size,
                              hipStream_t stream) {
    const float* q1 = (const float*)d_in[0];
    const float* q2 = (const float*)d_in[1];
    const float* q3 = (const float*)d_in[2];
    const float* s1 = (const float*)d_in[3];
    const float* s2 = (const float*)d_in[4];
    const float* s3 = (const float*)d_in[5];
    const float* r1 = (const float*)d_in[6];
    const float* r2 = (const float*)d_in[7];
    const float* r3 = (const float*)d_in[8];
    const int*   sm = (const int*)d_in[9];
    float* out = (float*)d_out;
    float* ws  = (float*)d_ws;

    k_resize<<<dim3((43008 + 255) / 256), 256, 0, stream>>>(sm, ws);
    k_counts<<<dim3(8, 3), 256, 0, stream>>>(ws);
    k_protos<<<dim3(256, 8, 3), 256, 0, stream>>>(s1, s2, s3, ws);
    k_build<<<dim3(8, 3), 256, 0, stream>>>(r1, r2, r3, ws);
    k_transform<<<dim3(128, 8, 2), 256, 0, stream>>>(q1, s1, ws, out, 4096, 0, 0,    5376);
    k_transform<<<dim3(32,  8, 2), 256, 0, stream>>>(q2, s2, ws, out, 1024, 1, 4096, 9472);
    k_transform<<<dim3(8,   8, 2), 256, 0, stream>>>(q3, s3, ws, out, 256,  2, 5120, 10496);
    k_pool_t<<<dim3(256, 8, 3), 256, 0, stream>>>(out, ws);
    k_topk<<<8, 256, 0, stream>>>(out, ws);
}